// Loss_32306744000922
// MI455X (gfx1250) — compile-verified
//
#include <hip/hip_runtime.h>

// Problem constants (from reference): B=64, C=2, H=512, W=512, POS_WEIGHT=10
#define HW        (512 * 512)          // 262144 elements per (b, c) plane
#define HW4       (HW / 4)             // 65536 float4 chunks per plane (power of 2)
#define NBATCH    64
#define TOTAL_V4  (NBATCH * HW4)       // 4,194,304 float4 chunks of channel-0 data
#define NB_BLOCKS 4096
#define NTHREADS  256                  // 8 wave32s per block
#define NITER     (TOTAL_V4 / (NB_BLOCKS * NTHREADS))   // == 4, exact
#define FITER     (NB_BLOCKS / NTHREADS)                // == 16, exact

typedef __attribute__((ext_vector_type(2))) float v2f;
typedef __attribute__((ext_vector_type(4))) float v4f;
typedef __attribute__((ext_vector_type(8))) float v8f;

// Full 32-lane reduction using V_WMMA_F32_16X16X4_F32:
//   A: VGPR0 = p (lanes 0-15 -> A[m][0]=p[m]; lanes 16-31 -> A[m][2]=p[m+16]),
//      VGPR1 = 0.  B = ones(4x16).
//   D[m][n] = rowsum(A)[m] = p[m] + p[m+16].
//   Lanes 0-15 hold D rows 0-7, lanes 16-31 hold rows 8-15; summing own 8 D
//   regs + one xor-16 shuffle yields the full wave sum in every lane.
// EXEC must be all ones at the call site (callers guarantee this).
__device__ __forceinline__ float wave_reduce_wmma(float p) {
    v2f a;
    a[0] = p;
    a[1] = 0.0f;
    v2f b;
    b[0] = 1.0f;
    b[1] = 1.0f;
    v8f c = {};
    v8f d = __builtin_amdgcn_wmma_f32_16x16x4_f32(
        /*neg_a=*/false, a, /*neg_b=*/false, b,
        /*c_mod=*/(short)0, c, /*reuse_a=*/false, /*reuse_b=*/false);
    float s = ((d[0] + d[1]) + (d[2] + d[3])) + ((d[4] + d[5]) + (d[6] + d[7]));
    s += __shfl_xor(s, 16, 32);
    return s;
}

// Per-element weighted BCE term:
//   -(y*log(x)*10 + (1-y)*log(1-x))  ==  -(l1x + y*(10*lx - l1x))
// v_log_f32 is base-2; scale by ln(2). x is bounded in [0.01, 0.99] so
// log(1-x) is a safe substitute for log1p(-x) in f32.
__device__ __forceinline__ float bce_term(float xx, float yy) {
    const float LN2 = 0.69314718055994530942f;
    float lx  = LN2 * __builtin_amdgcn_logf(xx);
    float l1x = LN2 * __builtin_amdgcn_logf(1.0f - xx);
    return -(l1x + yy * (10.0f * lx - l1x));
}

__global__ void __launch_bounds__(NTHREADS)
bce_partial_kernel(const float* __restrict__ x, const float* __restrict__ y,
                   float* __restrict__ ws) {
    const v4f* __restrict__ x4 = (const v4f*)x;
    const v4f* __restrict__ y4 = (const v4f*)y;

    const unsigned tid = blockIdx.x * NTHREADS + threadIdx.x;

    // Exactly NITER uniform iterations: issue all loads first (non-temporal,
    // single-use streaming data), then do the transcendental-heavy math.
    v4f xv[NITER], yv[NITER];
    #pragma unroll
    for (int k = 0; k < NITER; ++k) {
        unsigned i = tid + (unsigned)k * (NB_BLOCKS * NTHREADS);
        unsigned b = i >> 16;               // i / HW4
        unsigned r = i & (HW4 - 1);         // i % HW4
        unsigned idx = (b << 17) + r;       // b * (2*HW4) + r  -> channel 0 slice
        xv[k] = __builtin_nontemporal_load(&x4[idx]);
        yv[k] = __builtin_nontemporal_load(&y4[idx]);
    }

    float acc = 0.0f;
    #pragma unroll
    for (int k = 0; k < NITER; ++k) {
        acc += bce_term(xv[k].x, yv[k].x);
        acc += bce_term(xv[k].y, yv[k].y);
        acc += bce_term(xv[k].z, yv[k].z);
        acc += bce_term(xv[k].w, yv[k].w);
    }

    // EXEC is all ones here (no divergence above).
    float s = wave_reduce_wmma(acc);

    __shared__ float wsum[NTHREADS / 32];
    const unsigned lane = threadIdx.x & 31u;
    const unsigned wid  = threadIdx.x >> 5;
    if (lane == 0) wsum[wid] = s;
    __syncthreads();
    if (threadIdx.x == 0) {
        float t = 0.0f;
        #pragma unroll
        for (int w = 0; w < NTHREADS / 32; ++w) t += wsum[w];
        ws[blockIdx.x] = t;   // every block writes exactly one slot: ws fully defined
    }
}

__global__ void __launch_bounds__(NTHREADS)
bce_final_kernel(const float* __restrict__ ws, float* __restrict__ out) {
    float acc = 0.0f;
    #pragma unroll
    for (int k = 0; k < FITER; ++k)
        acc += ws[threadIdx.x + (unsigned)k * NTHREADS];

    float s = wave_reduce_wmma(acc);

    __shared__ float wsum[NTHREADS / 32];
    const unsigned lane = threadIdx.x & 31u;
    const unsigned wid  = threadIdx.x >> 5;
    if (lane == 0) wsum[wid] = s;
    __syncthreads();
    if (threadIdx.x == 0) {
        float t = 0.0f;
        #pragma unroll
        for (int w = 0; w < NTHREADS / 32; ++w) t += wsum[w];
        out[0] = t;
    }
}

extern "C" void kernel_launch(void* const* d_in, const int* in_sizes, int n_in,
                              void* d_out, int out_size, void* d_ws, size_t ws_size,
                              hipStream_t stream) {
    const float* x = (const float*)d_in[0];   // (64, 2, 512, 512) f32
    const float* y = (const float*)d_in[1];   // (64, 2, 512, 512) f32
    float* out = (float*)d_out;               // 1 element f32
    float* ws  = (float*)d_ws;                // >= NB_BLOCKS floats (16 KB)

    bce_partial_kernel<<<NB_BLOCKS, NTHREADS, 0, stream>>>(x, y, ws);
    bce_final_kernel<<<1, NTHREADS, 0, stream>>>(ws, out);
}